// LIFNeuron_86397562126867
// MI455X (gfx1250) — compile-verified
//
#include <hip/hip_runtime.h>
#include <cstdint>

// LIF scan: x (64,100,4096) f32 -> spikes (64,100,4096) f32 + scalar rate.
// Streaming workload: ~210 MB @ 23.3 TB/s ~= 9 us floor; no matmul -> WMMA
// inapplicable. CDNA5 path: GLOBAL_LOAD_ASYNC_TO_LDS_B128 double-buffered on
// ASYNCcnt; ~20 MB kept in flight chip-wide (matches HBM bandwidth-delay
// product) with zero VGPR pressure. Each lane runs 4 independent LIF chains
// (b128 granularity) for ILP on the serial u-recurrence.

#define T_STEPS 100
#define FEATS   4096
#define BATCH   64
#define BLOCK   128                // threads per block
#define VEC     4                  // floats per lane (b128)
#define CHUNK   20                 // timesteps per LDS buffer
#define NCHUNK  (T_STEPS / CHUNK)  // 5
#define ROWB    (BLOCK * VEC * 4)  // bytes per t-row in LDS: 2048
#define LDSB    (2 * CHUNK * ROWB) // dynamic LDS bytes: 81920
#define NTOT    ((size_t)BATCH * T_STEPS * FEATS)

static_assert(CHUNK == 20, "s_wait_asynccnt literal below assumes CHUNK==20");
static_assert(T_STEPS % CHUNK == 0, "chunking must divide T");
static_assert(FEATS % (BLOCK * VEC) == 0, "block must tile features");

typedef float v4f __attribute__((ext_vector_type(4)));

__global__ __launch_bounds__(BLOCK) void lif_scan_kernel(
    const float* __restrict__ x, const float* __restrict__ log_alpha,
    float* __restrict__ spikes, unsigned* __restrict__ count)
{
    extern __shared__ v4f smem[];  // [2][CHUNK][BLOCK] v4f = 80 KB

    const int tid    = threadIdx.x;
    const int fTiles = FEATS / (BLOCK * VEC);          // 8
    const int b      = blockIdx.x / fTiles;
    const int f0     = (blockIdx.x % fTiles) * (BLOCK * VEC) + tid * VEC;

    const float la    = log_alpha[0];
    const float alpha = 1.0f / (1.0f + __expf(-la));   // sigmoid(log_alpha)

    const float* xbase = x      + (size_t)b * T_STEPS * FEATS + f0; // 16B aligned
    v4f*         optr  = (v4f*)(spikes + (size_t)b * T_STEPS * FEATS + f0);

    // LDS byte offset of this lane's v4f slot (low 32 bits of flat shared ptr).
    const uint32_t lds0 = (uint32_t)(uintptr_t)(&smem[tid]);
    // Running global cursor; issue order is strictly t-ascending.
    uint64_t gaddr = (uint64_t)(uintptr_t)xbase;

    auto issue_chunk = [&](int c) {
        uint32_t laddr = lds0 + (uint32_t)(c & 1) * (CHUNK * ROWB);
        #pragma unroll
        for (int tt = 0; tt < CHUNK; ++tt) {
            // Lane DMAs 16B (4 features) of x[b, t, f0..f0+3] into its LDS slot.
            // 512 B per wave per instruction; tracked on ASYNCcnt, no VGPRs held.
            asm volatile("global_load_async_to_lds_b128 %0, %1, off"
                         :: "v"(laddr), "v"(gaddr)
                         : "memory");
            laddr += ROWB;                 // next t-row in this buffer
            gaddr += (uint64_t)FEATS * 4;  // next t in global (16 KB stride)
        }
    };

    v4f u   = {0.0f, 0.0f, 0.0f, 0.0f};
    int cnt = 0;

    issue_chunk(0);
    for (int c = 0; c < NCHUNK; ++c) {
        if (c + 1 < NCHUNK) {
            issue_chunk(c + 1);
            // <=20 outstanding (chunk c+1); async loads complete in issue
            // order, so chunk c is fully resident in LDS.
            asm volatile("s_wait_asynccnt 20" ::: "memory");
        } else {
            asm volatile("s_wait_asynccnt 0" ::: "memory");
        }

        const v4f* sptr = &smem[(c & 1) * CHUNK * BLOCK + tid];
        #pragma unroll
        for (int tt = 0; tt < CHUNK; ++tt) {
            v4f xv = sptr[tt * BLOCK];              // ds_load_b128, own slot
            u = alpha * u + xv;                     // 4 leaky-integrate chains
            bool f0b = (u.x >= 1.0f), f1b = (u.y >= 1.0f);
            bool f2b = (u.z >= 1.0f), f3b = (u.w >= 1.0f);
            v4f s = { f0b ? 1.0f : 0.0f, f1b ? 1.0f : 0.0f,
                      f2b ? 1.0f : 0.0f, f3b ? 1.0f : 0.0f };
            u -= s;                                 // subtract-reset (theta=1)
            cnt += (int)f0b + (int)f1b + (int)f2b + (int)f3b;
            __builtin_nontemporal_store(s, optr);   // b128 NT store, write-once
            optr += FEATS / VEC;                    // next t row
        }
    }

    // Exact integer reduction (deterministic across graph replays).
    __syncthreads();                  // all waves done with their LDS slots
    int* r = (int*)smem;
    r[tid] = cnt;
    __syncthreads();
    #pragma unroll
    for (int off = BLOCK / 2; off > 0; off >>= 1) {
        if (tid < off) r[tid] += r[tid + off];
        __syncthreads();
    }
    if (tid == 0) atomicAdd(count, (unsigned)r[0]);
}

__global__ void lif_zero_kernel(unsigned* __restrict__ count) {
    if (threadIdx.x == 0 && blockIdx.x == 0) *count = 0u;
}

__global__ void lif_rate_kernel(const unsigned* __restrict__ count,
                                float* __restrict__ rate_out) {
    if (threadIdx.x == 0 && blockIdx.x == 0) {
        rate_out[0] = (float)(*count) * (1.0f / (float)NTOT);
    }
}

extern "C" void kernel_launch(void* const* d_in, const int* in_sizes, int n_in,
                              void* d_out, int out_size, void* d_ws, size_t ws_size,
                              hipStream_t stream) {
    const float* x         = (const float*)d_in[0];   // (64,100,4096) f32
    const float* log_alpha = (const float*)d_in[1];   // scalar f32
    float*       out       = (float*)d_out;           // spikes flat + rate
    unsigned*    count     = (unsigned*)d_ws;         // scratch spike counter

    lif_zero_kernel<<<1, 1, 0, stream>>>(count);

    const int nblocks = BATCH * (FEATS / (BLOCK * VEC));  // 512
    lif_scan_kernel<<<nblocks, BLOCK, LDSB, stream>>>(x, log_alpha, out, count);

    lif_rate_kernel<<<1, 1, 0, stream>>>(count, out + NTOT);
}